// Decoder_57956288692356
// MI455X (gfx1250) — compile-verified
//
#include <hip/hip_runtime.h>
#include <math.h>

typedef __attribute__((ext_vector_type(2))) float v2f;
typedef __attribute__((ext_vector_type(8))) float v8f;

#define LDX 132          // padded row stride for concat(x, pos) buffer (multiple of 4)
#define KTILE 2048       // knn candidate tile (24KB LDS)
#define FLT_BIG 3.402823466e38f

// ---------- order-preserving float <-> uint encoding for atomicMax ----------
__device__ __forceinline__ unsigned fenc(float f) {
  unsigned u = __float_as_uint(f);
  return (u & 0x80000000u) ? ~u : (u | 0x80000000u);
}
__device__ __forceinline__ float fdec(unsigned e) {
  unsigned u = (e & 0x80000000u) ? (e & 0x7FFFFFFFu) : ~e;
  return __uint_as_float(u);
}

// ---------- concat(features, pos) -> xbuf[n, LDX], zero-padded cols ----------
__global__ void k_concat(const float* __restrict__ f, int fd,
                         const float* __restrict__ pos, int n,
                         float* __restrict__ xb) {
  int i = blockIdx.x * blockDim.x + threadIdx.x;
  if (i >= n * LDX) return;
  int r = i / LDX, c = i % LDX;
  float v = 0.f;
  if (c < fd)           v = f[r * fd + c];
  else if (c < fd + 3)  v = pos[r * 3 + (c - fd)];
  xb[i] = v;
}

// ---------- stage W[ci,co] into zero-padded pair-packed Wp ----------
// Wp[((k>>1)*co + c)*2 + (k&1)] = (k < ci) ? W[k*co+c] : 0   for k in [0, kpad)
__global__ void k_padw(const float* __restrict__ W, int ci, int co, int kpad,
                       float* __restrict__ Wp) {
  int i = blockIdx.x * blockDim.x + threadIdx.x;
  if (i >= kpad * co) return;
  int k = i / co, c = i % co;
  float v = (k < ci) ? W[k * co + c] : 0.f;
  Wp[((k >> 1) * co + c) * 2 + (k & 1)] = v;
}

// ---------- Y[n,co] = X[n,*] @ W  via V_WMMA_F32_16X16X4_F32 ----------
// 8 waves per block, one 16x16 output tile per wave. Branch-free K loop:
// A pair and B pair are each one global_load_b64; M/N clamped (not guarded).
__global__ void k_gemm_wmma(const float* __restrict__ X, int n, int kpad,
                            const float* __restrict__ Wp, int co,
                            float* __restrict__ Y) {
  int lane = threadIdx.x & 31;
  int wave = threadIdx.x >> 5;
  int tilesN = (co + 15) >> 4;
  int tilesM = (n + 15) >> 4;
  int tile = blockIdx.x * 8 + wave;
  if (tile >= tilesM * tilesN) return;   // wave-uniform: EXEC all-ones at WMMA
  int tm = tile / tilesN, tn = tile % tilesN;
  int half = lane >> 4, l15 = lane & 15;
  int row  = min(tm * 16 + l15, n - 1);   // clamped: rows >= n never stored
  int col  = tn * 16 + l15;
  int colc = min(col, co - 1);            // clamped: cols >= co never stored
  const float* ap = X + row * LDX + half * 2;          // (K=k0+2h, K=k0+2h+1)
  const float* bp = Wp + (half * co + colc) * 2;       // packed pair, same Ks
  v8f acc = {0.f, 0.f, 0.f, 0.f, 0.f, 0.f, 0.f, 0.f};
  for (int k0 = 0; k0 < kpad; k0 += 4) {
    v2f a = *(const v2f*)ap;
    v2f b = *(const v2f*)bp;
    acc = __builtin_amdgcn_wmma_f32_16x16x4_f32(false, a, false, b,
                                                (short)0, acc, false, false);
    ap += 4;
    bp += 4 * co;   // k0/2 advances by 2 -> packed offset advances 2*co*2
  }
  if (col < co) {
    int rb = tm * 16 + half * 8;         // C/D layout: half selects M or M+8
#pragma unroll
    for (int v = 0; v < 8; ++v) {
      int r = rb + v;
      if (r < n) Y[r * co + col] = acc[v];
    }
  }
}

// ---------- per-edge degree + mean edge-attr accumulation ----------
__global__ void k_edge_prep(const int* __restrict__ ei, int E,
                            const float* __restrict__ pos,
                            float* __restrict__ deg, float* __restrict__ mea) {
  int e = blockIdx.x * blockDim.x + threadIdx.x;
  if (e >= E) return;
  int s = ei[e], d = ei[E + e];
  float ex = pos[d * 3 + 0] - pos[s * 3 + 0];
  float ey = pos[d * 3 + 1] - pos[s * 3 + 1];
  float ez = pos[d * 3 + 2] - pos[s * 3 + 2];
  atomicAdd(&deg[d], 1.f);
  atomicAdd(&mea[d * 4 + 0], ex);
  atomicAdd(&mea[d * 4 + 1], ey);
  atomicAdd(&mea[d * 4 + 2], ez);
}

__global__ void k_mean_div(float* __restrict__ mea, const float* __restrict__ deg, int n) {
  int i = blockIdx.x * blockDim.x + threadIdx.x;
  if (i >= n) return;
  float inv = 1.f / fmaxf(deg[i], 1.f);
  mea[i * 4 + 0] *= inv;
  mea[i * 4 + 1] *= inv;
  mea[i * 4 + 2] *= inv;
}

// ---------- one wave per edge: logits + segment max ----------
__global__ void k_edge_logits(const int* __restrict__ ei, int E,
                              const float* __restrict__ pos,
                              const float* __restrict__ xl, const float* __restrict__ xr,
                              const float* __restrict__ We, const float* __restrict__ att,
                              int co, float* __restrict__ logits,
                              unsigned* __restrict__ lmax) {
  int wid = (blockIdx.x * blockDim.x + threadIdx.x) >> 5;
  int lane = threadIdx.x & 31;
  if (wid >= E) return;
  int s = ei[wid], d = ei[E + wid];
  float ex = pos[d * 3 + 0] - pos[s * 3 + 0];
  float ey = pos[d * 3 + 1] - pos[s * 3 + 1];
  float ez = pos[d * 3 + 2] - pos[s * 3 + 2];
  float part = 0.f;
  for (int c = lane; c < co; c += 32) {
    float t = xl[s * co + c] + xr[d * co + c]
            + ex * We[c] + ey * We[co + c] + ez * We[2 * co + c];
    t = (t > 0.f) ? t : 0.2f * t;        // leaky_relu(0.2)
    part += t * att[c];
  }
#pragma unroll
  for (int off = 16; off; off >>= 1) part += __shfl_xor(part, off, 32);
  if (lane == 0) {
    logits[wid] = part;
    atomicMax(&lmax[d], fenc(part));
  }
}

// ---------- one wave per node: self-loop logits ----------
__global__ void k_loop_logits(const float* __restrict__ mea,
                              const float* __restrict__ xl, const float* __restrict__ xr,
                              const float* __restrict__ We, const float* __restrict__ att,
                              int co, int n, float* __restrict__ loopl,
                              unsigned* __restrict__ lmax) {
  int wid = (blockIdx.x * blockDim.x + threadIdx.x) >> 5;
  int lane = threadIdx.x & 31;
  if (wid >= n) return;
  float ex = mea[wid * 4 + 0], ey = mea[wid * 4 + 1], ez = mea[wid * 4 + 2];
  float part = 0.f;
  for (int c = lane; c < co; c += 32) {
    float t = xl[wid * co + c] + xr[wid * co + c]
            + ex * We[c] + ey * We[co + c] + ez * We[2 * co + c];
    t = (t > 0.f) ? t : 0.2f * t;
    part += t * att[c];
  }
#pragma unroll
  for (int off = 16; off; off >>= 1) part += __shfl_xor(part, off, 32);
  if (lane == 0) {
    loopl[wid] = part;
    atomicMax(&lmax[wid], fenc(part));
  }
}

// ---------- softmax: exp + segment sum ----------
__global__ void k_edge_exp(const int* __restrict__ ei, int E,
                           float* __restrict__ logits,
                           const unsigned* __restrict__ lmax,
                           float* __restrict__ denom) {
  int e = blockIdx.x * blockDim.x + threadIdx.x;
  if (e >= E) return;
  int d = ei[E + e];
  float v = expf(logits[e] - fdec(lmax[d]));
  logits[e] = v;
  atomicAdd(&denom[d], v);
}

__global__ void k_loop_exp(float* __restrict__ loopl,
                           const unsigned* __restrict__ lmax,
                           float* __restrict__ denom, int n) {
  int i = blockIdx.x * blockDim.x + threadIdx.x;
  if (i >= n) return;
  float v = expf(loopl[i] - fdec(lmax[i]));
  loopl[i] = v;
  denom[i] += v;   // exclusive per node; edge sums already complete
}

// ---------- one wave per edge: weighted scatter aggregate ----------
__global__ void k_edge_aggr(const int* __restrict__ ei, int E,
                            const float* __restrict__ elog,
                            const float* __restrict__ denom,
                            const float* __restrict__ xl, int co,
                            float* __restrict__ out) {
  int wid = (blockIdx.x * blockDim.x + threadIdx.x) >> 5;
  int lane = threadIdx.x & 31;
  if (wid >= E) return;
  int s = ei[wid], d = ei[E + wid];
  float a = elog[wid] / denom[d];
  for (int c = lane; c < co; c += 32)
    atomicAdd(&out[d * co + c], a * xl[s * co + c]);
}

// ---------- self-loop contribution + bias + optional ELU ----------
__global__ void k_finalize(float* __restrict__ out,
                           const float* __restrict__ loopl,
                           const float* __restrict__ denom,
                           const float* __restrict__ xl,
                           const float* __restrict__ b,
                           int n, int co, int elu) {
  int idx = blockIdx.x * blockDim.x + threadIdx.x;
  if (idx >= n * co) return;
  int i = idx / co, c = idx % co;
  float a = loopl[i] / denom[i];
  float v = out[idx] + a * xl[i * co + c] + b[c];
  if (elu) v = (v > 0.f) ? v : expm1f(v);
  out[idx] = v;
}

// ---------- brute-force k=3 NN interpolation (LDS-tiled candidates) ----------
__device__ __forceinline__ void top3_ins(float& d0, float& d1, float& d2,
                                         int& j0, int& j1, int& j2,
                                         float x, int xi) {
  if (x < d2) {
    if (x < d1) {
      d2 = d1; j2 = j1;
      if (x < d0) { d1 = d0; j1 = j0; d0 = x; j0 = xi; }
      else        { d1 = x;  j1 = xi; }
    } else { d2 = x; j2 = xi; }
  }
}

__global__ void k_knn(const float* __restrict__ fx, int C,
                      const float* __restrict__ px, int nx,
                      const float* __restrict__ py, int ny,
                      float* __restrict__ out) {
  __shared__ float sp[KTILE * 3];
  int lane = threadIdx.x & 31;
  int wave = threadIdx.x >> 5;
  int q = blockIdx.x * 8 + wave;
  bool valid = (q < ny);
  float qx = 0.f, qy = 0.f, qz = 0.f;
  if (valid) { qx = py[q * 3]; qy = py[q * 3 + 1]; qz = py[q * 3 + 2]; }
  float d0 = FLT_BIG, d1 = FLT_BIG, d2 = FLT_BIG;
  int j0 = 0, j1 = 0, j2 = 0;
  for (int base = 0; base < nx; base += KTILE) {
    int cnt = min(KTILE, nx - base);
    __syncthreads();
    for (int t = threadIdx.x; t < cnt * 3; t += blockDim.x)
      sp[t] = px[base * 3 + t];
    __syncthreads();
    if (valid) {
      for (int j = lane; j < cnt; j += 32) {
        float ax = sp[j * 3] - qx, ay = sp[j * 3 + 1] - qy, az = sp[j * 3 + 2] - qz;
        float dd = ax * ax + ay * ay + az * az;
        top3_ins(d0, d1, d2, j0, j1, j2, dd, base + j);
      }
    }
  }
  // butterfly merge: disjoint coverage per stage -> all lanes hold global top-3
#pragma unroll
  for (int off = 16; off; off >>= 1) {
    float e0 = __shfl_xor(d0, off, 32), e1 = __shfl_xor(d1, off, 32), e2 = __shfl_xor(d2, off, 32);
    int   g0 = __shfl_xor(j0, off, 32), g1 = __shfl_xor(j1, off, 32), g2 = __shfl_xor(j2, off, 32);
    top3_ins(d0, d1, d2, j0, j1, j2, e0, g0);
    top3_ins(d0, d1, d2, j0, j1, j2, e1, g1);
    top3_ins(d0, d1, d2, j0, j1, j2, e2, g2);
  }
  if (!valid) return;
  float w0 = 1.f / fmaxf(d0, 1e-16f);
  float w1 = 1.f / fmaxf(d1, 1e-16f);
  float w2 = 1.f / fmaxf(d2, 1e-16f);
  float ws = w0 + w1 + w2;
  for (int c = lane; c < C; c += 32)
    out[q * C + c] = (w0 * fx[j0 * C + c] + w1 * fx[j1 * C + c] + w2 * fx[j2 * C + c]) / ws;
}

// =================== host orchestration ===================
struct GatWS {
  float* xbuf; float* xl; float* xr; float* feat; float* feat2;
  float* deg; float* mea; unsigned* lmax; float* denom; float* loopl; float* elog;
  float* wlp; float* wrp;
};

static void gat_layer(hipStream_t st, GatWS& w, const float* pos, const int* ei,
                      int E, int n, int ci,
                      const float* Wl, const float* Wr, const float* We,
                      const float* att, const float* b,
                      int co, float* outfeat, int elu) {
  const int T = 256;
  hipMemsetAsync(w.deg,   0, (size_t)n * sizeof(float), st);
  hipMemsetAsync(w.mea,   0, (size_t)n * 4 * sizeof(float), st);
  hipMemsetAsync(w.lmax,  0, (size_t)n * sizeof(unsigned), st);
  hipMemsetAsync(w.denom, 0, (size_t)n * sizeof(float), st);
  hipMemsetAsync(outfeat, 0, (size_t)n * co * sizeof(float), st);
  k_edge_prep<<<(E + T - 1) / T, T, 0, st>>>(ei, E, pos, w.deg, w.mea);
  k_mean_div<<<(n + T - 1) / T, T, 0, st>>>(w.mea, w.deg, n);
  int kpad = (ci + 3) & ~3;
  k_padw<<<((kpad * co) + T - 1) / T, T, 0, st>>>(Wl, ci, co, kpad, w.wlp);
  k_padw<<<((kpad * co) + T - 1) / T, T, 0, st>>>(Wr, ci, co, kpad, w.wrp);
  int tiles = ((n + 15) / 16) * ((co + 15) / 16);
  k_gemm_wmma<<<(tiles + 7) / 8, 256, 0, st>>>(w.xbuf, n, kpad, w.wlp, co, w.xl);
  k_gemm_wmma<<<(tiles + 7) / 8, 256, 0, st>>>(w.xbuf, n, kpad, w.wrp, co, w.xr);
  k_edge_logits<<<(E + 7) / 8, 256, 0, st>>>(ei, E, pos, w.xl, w.xr, We, att, co, w.elog, w.lmax);
  k_loop_logits<<<(n + 7) / 8, 256, 0, st>>>(w.mea, w.xl, w.xr, We, att, co, n, w.loopl, w.lmax);
  k_edge_exp<<<(E + T - 1) / T, T, 0, st>>>(ei, E, w.elog, w.lmax, w.denom);
  k_loop_exp<<<(n + T - 1) / T, T, 0, st>>>(w.loopl, w.lmax, w.denom, n);
  k_edge_aggr<<<(E + 7) / 8, 256, 0, st>>>(ei, E, w.elog, w.denom, w.xl, co, outfeat);
  k_finalize<<<((n * co) + T - 1) / T, T, 0, st>>>(outfeat, w.loopl, w.denom, w.xl, b, n, co, elu);
}

extern "C" void kernel_launch(void* const* d_in, const int* in_sizes, int n_in,
                              void* d_out, int out_size, void* d_ws, size_t ws_size,
                              hipStream_t stream) {
  (void)in_sizes; (void)n_in; (void)out_size; (void)ws_size;
  const float* latent = (const float*)d_in[0];
  const float* pos0   = (const float*)d_in[1];
  const float* pos1   = (const float*)d_in[2];
  const float* pos2   = (const float*)d_in[3];
  const int*   ei0    = (const int*)d_in[4];
  const int*   ei1    = (const int*)d_in[5];
  const int*   ei2    = (const int*)d_in[6];
  const float* P[5][5];
  for (int l = 0; l < 5; ++l)
    for (int j = 0; j < 5; ++j)
      P[l][j] = (const float*)d_in[7 + 5 * l + j];

  const int NMAX = 50000, EMAX = 800000, CO = 128;
  size_t off = 0;
  auto carve = [&](size_t bytes) {
    void* p = (char*)d_ws + off;
    off = (off + bytes + 255) & ~(size_t)255;
    return p;
  };
  GatWS w;
  w.xbuf  = (float*)carve((size_t)NMAX * LDX * 4);
  w.xl    = (float*)carve((size_t)NMAX * CO * 4);
  w.xr    = (float*)carve((size_t)NMAX * CO * 4);
  w.feat  = (float*)carve((size_t)NMAX * CO * 4);
  w.feat2 = (float*)carve((size_t)NMAX * CO * 4);
  w.deg   = (float*)carve((size_t)NMAX * 4);
  w.mea   = (float*)carve((size_t)NMAX * 16);
  w.lmax  = (unsigned*)carve((size_t)NMAX * 4);
  w.denom = (float*)carve((size_t)NMAX * 4);
  w.loopl = (float*)carve((size_t)NMAX * 4);
  w.elog  = (float*)carve((size_t)EMAX * 4);
  w.wlp   = (float*)carve((size_t)LDX * CO * 4);
  w.wrp   = (float*)carve((size_t)LDX * CO * 4);

  const int T = 256;
  // Layer 0: n=12500, ci=64+3
  k_concat<<<((12500 * LDX) + T - 1) / T, T, 0, stream>>>(latent, 64, pos0, 12500, w.xbuf);
  gat_layer(stream, w, pos0, ei0, 200000, 12500, 67,
            P[0][0], P[0][1], P[0][2], P[0][3], P[0][4], 128, w.feat, 1);
  // Layer 1: n=12500, ci=131
  k_concat<<<((12500 * LDX) + T - 1) / T, T, 0, stream>>>(w.feat, 128, pos0, 12500, w.xbuf);
  gat_layer(stream, w, pos0, ei0, 200000, 12500, 131,
            P[1][0], P[1][1], P[1][2], P[1][3], P[1][4], 128, w.feat, 1);
  // knn 0: level0 (12500) -> level1 (25000)
  k_knn<<<(25000 + 7) / 8, 256, 0, stream>>>(w.feat, 128, pos0, 12500, pos1, 25000, w.feat2);
  // Layer 2: n=25000
  k_concat<<<((25000 * LDX) + T - 1) / T, T, 0, stream>>>(w.feat2, 128, pos1, 25000, w.xbuf);
  gat_layer(stream, w, pos1, ei1, 400000, 25000, 131,
            P[2][0], P[2][1], P[2][2], P[2][3], P[2][4], 128, w.feat, 1);
  // knn 1: level1 (25000) -> level2 (50000)
  k_knn<<<(50000 + 7) / 8, 256, 0, stream>>>(w.feat, 128, pos1, 25000, pos2, 50000, w.feat2);
  // Layer 3: n=50000
  k_concat<<<((50000 * LDX) + T - 1) / T, T, 0, stream>>>(w.feat2, 128, pos2, 50000, w.xbuf);
  gat_layer(stream, w, pos2, ei2, 800000, 50000, 131,
            P[3][0], P[3][1], P[3][2], P[3][3], P[3][4], 128, w.feat, 1);
  // Layer 4: n=50000, co=5, no ELU, writes d_out
  k_concat<<<((50000 * LDX) + T - 1) / T, T, 0, stream>>>(w.feat, 128, pos2, 50000, w.xbuf);
  gat_layer(stream, w, pos2, ei2, 800000, 50000, 131,
            P[4][0], P[4][1], P[4][2], P[4][3], P[4][4], 5, (float*)d_out, 0);
}